// AbsBatchTopKActivation_27152783245523
// MI455X (gfx1250) — compile-verified
//
#include <hip/hip_runtime.h>
#include <stdint.h>

// ---------------------------------------------------------------------------
// AbsBatchTopK on MI455X (gfx1250): radix-threshold select on |x| bit pattern.
//   Pass 1: per-WGP 32768-bin LDS histogram of bits[30:16] of |x|
//           (128 KB LDS hist + 16 KB double-buffered async-to-LDS stage,
//            possible only with CDNA5's 320 KB LDS). Loads use default RT
//            temporal hints so x populates the 192 MB L2.
//   Pass 2: tiny scan -> threshold bin B, residual count R.
//   Pass 3: streaming apply; x re-read hits L2 (~75% of 256 MB resident),
//           y written with NT hints so the write-once output never evicts x.
//   Exact-K via rare-path global ticket on threshold-bin ties.
// ---------------------------------------------------------------------------

#define D_SAE      16384
#define HIST_BITS  15
#define HIST_BINS  (1 << HIST_BITS)     // bins over bits [30:16] of |x|
#define HIST_WGS   512
#define HIST_TPB   512                  // 16 wave32 per WGP
#define SCAN_TPB   1024
#define APPLY_TPB  256
#define APPLY_WGS  4096

typedef float    v4f __attribute__((ext_vector_type(4)));

// --- CDNA5 async-copy (global -> LDS), tracked by ASYNCcnt ------------------
__device__ __forceinline__ void async_load_b128_to_lds(const void* gptr,
                                                       uint32_t lds_off) {
  // VDST = VGPR holding LDS byte offset, VADDR = 64-bit global address (GV).
  asm volatile("global_load_async_to_lds_b128 %0, %1, off"
               :: "v"(lds_off), "v"(gptr)
               : "memory");
}

template <int N>
__device__ __forceinline__ void wait_asynccnt_le() {
#if __has_builtin(__builtin_amdgcn_s_wait_asynccnt)
  __builtin_amdgcn_s_wait_asynccnt(N);
#else
  asm volatile("s_wait_asynccnt %0" :: "i"(N) : "memory");
#endif
}

__device__ __forceinline__ uint32_t lds_offset_of(const void* p) {
  // Generic (flat) LDS address carries the byte offset in its low 32 bits.
  return (uint32_t)(uintptr_t)p;
}

// --- Kernel A: zero the global histogram -----------------------------------
__global__ void topk_zero_kernel(uint32_t* __restrict__ g) {
  g[(size_t)blockIdx.x * blockDim.x + threadIdx.x] = 0u;
}

// --- Kernel B: 32K-bin histogram of |x| prefix, async-staged through LDS ----
__global__ __launch_bounds__(HIST_TPB)
void topk_hist_kernel(const float4* __restrict__ xv,
                      uint32_t* __restrict__ ghist,
                      int iters) {
  __shared__ uint32_t hist[HIST_BINS];        // 128 KB
  __shared__ float4   stage[2][HIST_TPB];     // 16 KB, double-buffered

  const int tid = threadIdx.x;
  for (int b = tid; b < HIST_BINS; b += HIST_TPB) hist[b] = 0u;
  __syncthreads();

  // Each block owns a contiguous chunk: iters * HIST_TPB float4's.
  const float4* base = xv + (size_t)blockIdx.x * (size_t)iters * HIST_TPB;
  const uint32_t lds0 = lds_offset_of(&stage[0][tid]);
  const uint32_t lds1 = lds_offset_of(&stage[1][tid]);

  // Prime the pipeline.
  async_load_b128_to_lds(base + tid, lds0);

  for (int i = 0; i < iters; ++i) {
    if (i + 1 < iters) {
      async_load_b128_to_lds(base + (size_t)(i + 1) * HIST_TPB + tid,
                             ((i + 1) & 1) ? lds1 : lds0);
      wait_asynccnt_le<1>();   // tile i complete (in-order), i+1 in flight
    } else {
      wait_asynccnt_le<0>();
    }
    const float4 v = stage[i & 1][tid];       // ds_load_b128
    const float c0 = v.x, c1 = v.y, c2 = v.z, c3 = v.w;
    atomicAdd(&hist[(__float_as_uint(c0) & 0x7fffffffu) >> (31 - HIST_BITS)], 1u);
    atomicAdd(&hist[(__float_as_uint(c1) & 0x7fffffffu) >> (31 - HIST_BITS)], 1u);
    atomicAdd(&hist[(__float_as_uint(c2) & 0x7fffffffu) >> (31 - HIST_BITS)], 1u);
    atomicAdd(&hist[(__float_as_uint(c3) & 0x7fffffffu) >> (31 - HIST_BITS)], 1u);
  }

  __syncthreads();
  // Sparse flush: only nonzero bins touch L2 atomics (coalesced across lanes).
  for (int b = tid; b < HIST_BINS; b += HIST_TPB) {
    const uint32_t c = hist[b];
    if (c) atomicAdd(&ghist[b], c);
  }
}

// --- Kernel C: find threshold bin B and residual count R (single block) -----
__global__ __launch_bounds__(SCAN_TPB)
void topk_scan_kernel(const uint32_t* __restrict__ ghist,
                      const int* __restrict__ topk,
                      int bsz,
                      uint32_t* __restrict__ params) {
  __shared__ uint32_t csum[SCAN_TPB];
  const int t = threadIdx.x;
  constexpr int BPT = HIST_BINS / SCAN_TPB;   // 32 bins per thread

  uint32_t s = 0;
  for (int j = 0; j < BPT; ++j) s += ghist[t * BPT + j];
  csum[t] = s;
  __syncthreads();

  if (t == 0) {
    const uint32_t K = (uint32_t)topk[0] * (uint32_t)bsz;
    uint32_t running = 0;
    int chunk = -1;
    for (int c = SCAN_TPB - 1; c >= 0; --c) {
      const uint32_t cs = csum[c];
      if (running + cs >= K) { chunk = c; break; }
      running += cs;
    }
    uint32_t B = 0;
    uint32_t rem = (K > running) ? (K - running) : 0u;
    if (chunk >= 0) {
      for (int j = BPT - 1; j >= 0; --j) {
        const uint32_t c = ghist[chunk * BPT + j];
        if (running + c >= K) {
          B = (uint32_t)(chunk * BPT + j);
          rem = K - running;
          break;
        }
        running += c;
      }
    }
    params[0] = B;
    params[1] = rem;
    params[2] = 0u;   // ticket counter for exact-K tie handling
  }
}

// --- Kernel D: streaming apply (read x via L2, NT-store masked x) -----------
__global__ __launch_bounds__(APPLY_TPB)
void topk_apply_kernel(const v4f* __restrict__ xv,
                       v4f* __restrict__ yv,
                       const uint32_t* __restrict__ params,
                       uint32_t* __restrict__ ticket,
                       int nvec) {
  const uint32_t B   = params[0];
  const uint32_t rem = params[1];
  const int stride = gridDim.x * blockDim.x;

  for (int idx = blockIdx.x * blockDim.x + threadIdx.x; idx < nvec;
       idx += stride) {
    if (idx + stride < nvec)
      __builtin_prefetch(&xv[idx + stride], 0, 0);   // global_prefetch_b8

    const v4f v = xv[idx];                            // RT load: hits L2
    v4f o;
    {
      const uint32_t b = (__float_as_uint(v.x) & 0x7fffffffu) >> (31 - HIST_BITS);
      const bool keep = (b > B) || (b == B && atomicAdd(ticket, 1u) < rem);
      o.x = keep ? v.x : 0.0f;
    }
    {
      const uint32_t b = (__float_as_uint(v.y) & 0x7fffffffu) >> (31 - HIST_BITS);
      const bool keep = (b > B) || (b == B && atomicAdd(ticket, 1u) < rem);
      o.y = keep ? v.y : 0.0f;
    }
    {
      const uint32_t b = (__float_as_uint(v.z) & 0x7fffffffu) >> (31 - HIST_BITS);
      const bool keep = (b > B) || (b == B && atomicAdd(ticket, 1u) < rem);
      o.z = keep ? v.z : 0.0f;
    }
    {
      const uint32_t b = (__float_as_uint(v.w) & 0x7fffffffu) >> (31 - HIST_BITS);
      const bool keep = (b > B) || (b == B && atomicAdd(ticket, 1u) < rem);
      o.w = keep ? v.w : 0.0f;
    }
    // Write-once output: non-temporal so it never evicts x from the 192MB L2.
    __builtin_nontemporal_store(o, &yv[idx]);
  }
}

// ---------------------------------------------------------------------------
extern "C" void kernel_launch(void* const* d_in, const int* in_sizes, int n_in,
                              void* d_out, int out_size, void* d_ws,
                              size_t ws_size, hipStream_t stream) {
  const float* x    = (const float*)d_in[0];
  const int*   topk = (const int*)d_in[1];
  const int N   = in_sizes[0];          // 4096 * 16384 = 2^26
  const int bsz = N / D_SAE;

  uint32_t* ghist  = (uint32_t*)d_ws;           // HIST_BINS counters
  uint32_t* params = ghist + HIST_BINS;         // [B, rem, ticket]

  // A: zero histogram (32768 dwords).
  topk_zero_kernel<<<HIST_BINS / 512, 512, 0, stream>>>(ghist);

  // B: histogram pass (each block: iters * HIST_TPB float4's, contiguous).
  const int iters = N / (HIST_WGS * HIST_TPB * 4);   // 64 for N = 2^26
  topk_hist_kernel<<<HIST_WGS, HIST_TPB, 0, stream>>>(
      (const float4*)x, ghist, iters);

  // C: threshold scan (also resets the ticket counter every call).
  topk_scan_kernel<<<1, SCAN_TPB, 0, stream>>>(ghist, topk, bsz, params);

  // D: apply mask.
  const int nvec = N / 4;
  topk_apply_kernel<<<APPLY_WGS, APPLY_TPB, 0, stream>>>(
      (const v4f*)x, (v4f*)d_out, params, params + 2, nvec);
}